// CustomAlignmentLoss_2826088481390
// MI455X (gfx1250) — compile-verified
//
#include <hip/hip_runtime.h>
#include <hip/hip_bf16.h>

typedef __attribute__((ext_vector_type(2))) float v2f;
typedef __attribute__((ext_vector_type(8))) float v8f;

#define BLOCK 256          // 8 waves of 32
#define ROWS_PER_WAVE 16
#define ROWS_PER_BLOCK 128 // 8 waves * 16 rows
#define CHUNK 2048         // y-points staged in LDS per iteration (32 KB SoA pairs)

// For each point in X[b] (rows), min over all points of Y[b] (cols) of squared
// L2 distance via V_WMMA_F32_16X16X4_F32:
//   A row i = (-2x0, -2x1, -2x2, 1)
//   B col j = ( y0,   y1,   y2,  |y_j|^2)
//   C[i][j] = |x_i|^2   (loop-invariant, built once via shuffles)
//   => D[i][j] = |x_i|^2 + |y_j|^2 - 2 x_i.y_j = d2
__global__ __launch_bounds__(BLOCK) void nn_min_kernel(
    const float* __restrict__ X, const float* __restrict__ Y,
    float* __restrict__ outmin, int N, int M)
{
    // Pair layout matches the WMMA B-operand register layout:
    //   lanes 0-15 load (K0,K1) = (y0,y1); lanes 16-31 load (K2,K3) = (y2,|y|^2)
    __shared__ float2 lBlo[CHUNK];
    __shared__ float2 lBhi[CHUNK];

    const int blocksPerBatch = N / ROWS_PER_BLOCK;
    const int b        = blockIdx.x / blocksPerBatch;
    const int rowBlock = blockIdx.x % blocksPerBatch;
    const int wave     = threadIdx.x >> 5;
    const int lane     = threadIdx.x & 31;
    const int half     = lane >> 4;            // 0 = low half, 1 = high half
    const int base_n   = rowBlock * ROWS_PER_BLOCK + wave * ROWS_PER_WAVE;

    // ---- A operand: lane (l&15) owns row base_n + (l&15).
    // lanes 0-15: VGPR0=K0, VGPR1=K1 ; lanes 16-31: VGPR0=K2, VGPR1=K3
    const int arow = base_n + (lane & 15);
    const float* xp = X + ((size_t)b * N + arow) * 3;
    const float x0 = xp[0], x1 = xp[1], x2c = xp[2];
    const float xn = x0 * x0 + x1 * x1 + x2c * x2c;
    v2f a;
    a.x = half ? -2.0f * x2c : -2.0f * x0;
    a.y = half ? 1.0f        : -2.0f * x1;

    // ---- C operand: vgpr r holds row r (lanes 0-15) / row r+8 (lanes 16-31).
    // Gather |x|^2 of those rows once via shuffles (source lanes 0..15 hold them).
    v8f cxn;
    #pragma unroll
    for (int r = 0; r < 8; ++r)
        cxn[r] = __shfl(xn, (half << 3) + r, 32);

    v8f acc;
    #pragma unroll
    for (int i = 0; i < 8; ++i) acc[i] = 3.0e38f;

    const float* Yb = Y + (size_t)b * M * 3;
    // Branchless per-lane B base pointer (address select, no exec divergence)
    const float2* bptr = half ? lBhi : lBlo;
    const int jlane = lane & 15;

    for (int cbase = 0; cbase < M; cbase += CHUNK) {
        __syncthreads();
        // Cooperative stage of CHUNK y-points into B-operand pair layout
        for (int p = threadIdx.x; p < CHUNK; p += BLOCK) {
            const float* yp = Yb + (size_t)(cbase + p) * 3;
            const float t0 = yp[0], t1 = yp[1], t2 = yp[2];
            lBlo[p] = make_float2(t0, t1);
            lBhi[p] = make_float2(t2, t0 * t0 + t1 * t1 + t2 * t2);
        }
        __syncthreads();

        #pragma unroll 4
        for (int toff = 0; toff < CHUNK; toff += 16) {
            const float2 t = bptr[toff + jlane];   // one ds_load_b64 per tile
            v2f bb; bb.x = t.x; bb.y = t.y;

            v8f d = __builtin_amdgcn_wmma_f32_16x16x4_f32(
                false, a, false, bb, (short)0, cxn, false, false);

            #pragma unroll
            for (int i = 0; i < 8; ++i) acc[i] = fminf(acc[i], d[i]);
        }
    }

    // Min across the 16 columns held by each 16-lane half (wave32: masks 1..8
    // stay inside each half). After this, all lanes in a half hold the row min.
    #pragma unroll
    for (int mask = 1; mask <= 8; mask <<= 1) {
        #pragma unroll
        for (int i = 0; i < 8; ++i)
            acc[i] = fminf(acc[i], __shfl_xor(acc[i], mask, 32));
    }

    // D layout: vgpr r holds row r (lanes 0-15) / row r+8 (lanes 16-31)
    if ((lane & 15) == 0) {
        const int rbase = base_n + half * 8;
        #pragma unroll
        for (int r = 0; r < 8; ++r)
            outmin[(size_t)b * N + rbase + r] = acc[r];
    }
}

__global__ __launch_bounds__(512) void chamfer_reduce_kernel(
    const float* __restrict__ minx, const float* __restrict__ miny,
    float* __restrict__ out, int nx, int ny)
{
    __shared__ float sx[512], sy[512];
    float ax = 0.0f, ay = 0.0f;
    for (int i = threadIdx.x; i < nx; i += 512) ax += minx[i];
    for (int i = threadIdx.x; i < ny; i += 512) ay += miny[i];
    sx[threadIdx.x] = ax; sy[threadIdx.x] = ay;
    __syncthreads();
    for (int s = 256; s > 0; s >>= 1) {
        if ((int)threadIdx.x < s) {
            sx[threadIdx.x] += sx[threadIdx.x + s];
            sy[threadIdx.x] += sy[threadIdx.x + s];
        }
        __syncthreads();
    }
    if (threadIdx.x == 0)
        out[0] = 0.01f * (sx[0] / (float)nx + sy[0] / (float)ny);
}

extern "C" void kernel_launch(void* const* d_in, const int* in_sizes, int n_in,
                              void* d_out, int out_size, void* d_ws, size_t ws_size,
                              hipStream_t stream) {
    (void)n_in; (void)out_size; (void)ws_size;
    const float* src = (const float*)d_in[0]; // [B, N, 3]
    const float* tgt = (const float*)d_in[1]; // [B, M, 3]
    const int B = 4, D = 3;
    const int N = in_sizes[0] / (B * D);
    const int M = in_sizes[1] / (B * D);

    float* minx = (float*)d_ws;               // B*N floats: min_m d2
    float* miny = minx + (size_t)B * N;       // B*M floats: min_n d2

    // Pass 1: source -> nearest target ; Pass 2: target -> nearest source
    nn_min_kernel<<<B * (N / ROWS_PER_BLOCK), BLOCK, 0, stream>>>(src, tgt, minx, N, M);
    nn_min_kernel<<<B * (M / ROWS_PER_BLOCK), BLOCK, 0, stream>>>(tgt, src, miny, M, N);
    chamfer_reduce_kernel<<<1, 512, 0, stream>>>(minx, miny, (float*)d_out, B * N, B * M);
}